// RandomShift_33079838113837
// MI455X (gfx1250) — compile-verified
//
#include <hip/hip_runtime.h>

// RandomShift: out[row, t] = x[row, t - s],  s = shifts[row] - MAX_SHIFT, zero outside [0,T)
// 0 FLOPs, ~328 MB HBM traffic -> pure bandwidth kernel (~14us floor @ 23.3 TB/s).
// Strategy: shift s is uniform per row => alignment residue r = (-s)&3 and the
// interior/boundary test are SCALAR (SGPR) decisions. Hot path: aligned b128 NT
// loads/stores, 4 vec4 per thread (5 loads : 4 stores), static register shuffle.

typedef __attribute__((ext_vector_type(4))) float v4f;

constexpr int TLEN   = 160000;        // T
constexpr int MAXS   = 16000;         // T * 0.1
constexpr int NROWS  = 32 * 8;        // B * M
constexpr int BLOCK  = 256;           // 8 wave32 waves
constexpr int ITEMS  = 4;             // vec4s per thread (64B out per lane)
constexpr int CHUNK  = 4 * ITEMS;     // 16 floats per thread
constexpr int TC     = TLEN / CHUNK;  // 10000 thread-chunks per row (exact)
constexpr int PF_ELEMS = 8192;        // prefetch 32KB ahead in source stream

// Fast interior path: R = (-s)&3 is a compile-time constant per instantiation,
// selected by a scalar branch. All loads/stores aligned b128, non-temporal.
template <int R>
__device__ __forceinline__ void fast_chunk(const float* __restrict__ xr,
                                           float* __restrict__ outr,
                                           int t0, int a)
{
    constexpr int NLOAD = ITEMS + (R ? 1 : 0);
    v4f w[ITEMS + 1];
    #pragma unroll
    for (int k = 0; k < NLOAD; ++k)
        w[k] = __builtin_nontemporal_load((const v4f*)(xr + a + 4 * k));
    #pragma unroll
    for (int k = 0; k < ITEMS; ++k) {
        v4f o;
        #pragma unroll
        for (int j = 0; j < 4; ++j) {
            constexpr int dummy = 0; (void)dummy;
            const int idx = R + j;                 // compile-time within unroll
            o[j] = (idx < 4) ? w[k][idx] : w[k + 1][idx - 4];
        }
        __builtin_nontemporal_store(o, (v4f*)(outr + t0 + 4 * k));
    }
}

__global__ __launch_bounds__(BLOCK) void RandomShift_kernel(
    const float* __restrict__ x,
    const int*   __restrict__ shifts,
    float*       __restrict__ out)
{
    const int row = blockIdx.y;                 // uniform
    const int s   = shifts[row] - MAXS;         // s_load_b32 -> SGPR, in [-16000,16000]
    const int r   = (-s) & 3;                   // SGPR: residue of q = t0 - s (t0 % 4 == 0)

    const size_t rowOff = (size_t)row * (size_t)TLEN;
    const float* __restrict__ xr   = x   + rowOff;
    float*       __restrict__ outr = out + rowOff;

    const int tc = blockIdx.x * BLOCK + threadIdx.x;   // chunk index within row
    const int t0 = tc * CHUNK;                         // first output element (16B aligned)
    const int a  = t0 - s - r;                         // aligned source base for this chunk

    // ---- Block-uniform interior test (all SALU: blockIdx + s only) ----
    const int blkT0   = blockIdx.x * (BLOCK * CHUNK);
    const int aFirst  = blkT0 - s - r;                           // min load addr in block
    const int aLast   = blkT0 + (BLOCK - 1) * CHUNK - s - r;     // last chunk's base
    const bool blockInterior =
        (aFirst >= 0) &&
        (aLast + 4 * (ITEMS + 1) <= TLEN) &&                     // max load end (conservative)
        (blkT0 + BLOCK * CHUNK <= TLEN);                         // all stores in-range

    if (blockInterior) {
        // gfx1250 speculative prefetch of the source stream (OOB dropped silently).
        __builtin_prefetch(xr + a + PF_ELEMS, 0, 0);
        switch (r) {                                 // scalar 4-way branch (s_cbranch)
            case 0:  fast_chunk<0>(xr, outr, t0, a); break;
            case 1:  fast_chunk<1>(xr, outr, t0, a); break;
            case 2:  fast_chunk<2>(xr, outr, t0, a); break;
            default: fast_chunk<3>(xr, outr, t0, a); break;
        }
    } else {
        // Boundary blocks (<= ~5 of 40 per row): guarded per-element path.
        if (tc >= TC) return;
        #pragma unroll
        for (int k = 0; k < ITEMS; ++k) {
            const int tb = t0 + 4 * k;
            v4f o;
            #pragma unroll
            for (int j = 0; j < 4; ++j) {
                const int src = tb + j - s;
                o[j] = (src >= 0 && src < TLEN) ? xr[src] : 0.0f;
            }
            __builtin_nontemporal_store(o, (v4f*)(outr + tb));
        }
    }
}

extern "C" void kernel_launch(void* const* d_in, const int* in_sizes, int n_in,
                              void* d_out, int out_size, void* d_ws, size_t ws_size,
                              hipStream_t stream)
{
    const float* x      = (const float*)d_in[0];   // [32, 8, 160000] fp32
    const int*   shifts = (const int*)d_in[1];     // [32, 8] int32
    float*       out    = (float*)d_out;           // [32, 8, 160000] fp32

    dim3 block(BLOCK, 1, 1);
    dim3 grid((TC + BLOCK - 1) / BLOCK, NROWS, 1); // 40 x 256 blocks, 8 waves each

    RandomShift_kernel<<<grid, block, 0, stream>>>(x, shifts, out);
}